// GwPFM_72670846648706
// MI455X (gfx1250) — compile-verified
//
#include <hip/hip_runtime.h>
#include <stdint.h>

// Problem constants (match reference)
#define B_TOT   2048
#define F       32
#define G       8
#define D       64

// Tiling
#define NB              4                      // b-values per block
#define THREADS         256                    // 8 waves (wave32)
#define CHUNK_F         4                      // fields per pipeline stage
#define NCHUNK          (F / CHUNK_F)          // 8 stages
#define SPAN_FLOATS     (CHUNK_F * G * D)      // per-b contiguous span per chunk = 2048 floats
#define SPAN_F4         (SPAN_FLOATS / 4)      // 512 float4
#define CHUNK_FLOATS    (NB * SPAN_FLOATS)     // 8192 floats = 32 KB per chunk
#define CHUNK_F4        (CHUNK_FLOATS / 4)     // 2048 float4
#define COPIES_PER_THR  (CHUNK_F4 / THREADS)   // 8 async b128 per thread (== per wave) per chunk
#define FGD             (F * G * D)            // floats per b = 16384

// CDNA5 async global -> LDS copy, 16B per lane, tracked by ASYNCcnt
__device__ __forceinline__ void async_copy_b128(uint32_t lds_byte_off, const float* gptr) {
    asm volatile("global_load_async_to_lds_b128 %0, %1, off"
                 :: "v"(lds_byte_off), "v"(gptr)
                 : "memory");
}
#define ASYNC_WAIT(N) asm volatile("s_wait_asynccnt " #N ::: "memory")

__global__ __launch_bounds__(THREADS)
void GwPFM_pairsum_kernel(const float* __restrict__ inputs,
                          const float* __restrict__ corr,
                          float* __restrict__ out) {
    __shared__ __align__(16) float tile[2 * CHUNK_FLOATS];  // 64 KB double buffer
    __shared__ float wsh[G * G];                            // 64 weights

    const int tid = threadIdx.x;
    const int b0  = blockIdx.x * NB;

    if (tid < G * G) wsh[tid] = corr[tid];   // correlation[gi][gj][0]

    const float*   gblock    = inputs + (size_t)b0 * FGD;
    const uint32_t tile_base = (uint32_t)(uintptr_t)(void*)tile;  // wave-relative LDS byte offset

    // Issue one pipeline-stage worth of async copies (jc's 4-field slab for all NB b's)
    auto issue_chunk = [&](int jc, int buf) {
        const float*   gchunk = gblock + jc * SPAN_FLOATS;
        const uint32_t lbuf   = tile_base + (uint32_t)buf * (CHUNK_FLOATS * 4u);
        #pragma unroll
        for (int it = 0; it < COPIES_PER_THR; ++it) {
            int c   = it * THREADS + tid;           // float4 id in [0, 2048)
            int b_l = c / SPAN_F4;
            int rem = c - b_l * SPAN_F4;            // float4 within the 8KB per-b span
            const float* g = gchunk + (size_t)b_l * FGD + rem * 4;
            uint32_t     l = lbuf + (uint32_t)(b_l * SPAN_FLOATS + rem * 4) * 4u;
            async_copy_b128(l, g);
        }
    };

    // Per-thread state: running prefix sums S[gi][g] (compile-time indexed -> registers)
    float S[G][G];
    #pragma unroll
    for (int a = 0; a < G; ++a)
        #pragma unroll
        for (int b = 0; b < G; ++b) S[a][b] = 0.0f;
    float acc = 0.0f;

    const int q = tid >> 6;      // b_local 0..3
    const int d = tid & 63;      // output lane, contiguous in memory

    issue_chunk(0, 0);

    #pragma unroll
    for (int jc = 0; jc < NCHUNK; ++jc) {
        if (jc + 1 < NCHUNK) {
            issue_chunk(jc + 1, (jc + 1) & 1);
            ASYNC_WAIT(8);                 // chunk jc done; chunk jc+1 still in flight
        } else {
            ASYNC_WAIT(0);                 // last chunk: drain
        }
        __syncthreads();                   // cross-wave visibility of LDS tile

        const float* lt = &tile[((jc & 1) * CHUNK_FLOATS) + q * SPAN_FLOATS + d];

        #pragma unroll
        for (int jl = 0; jl < CHUNK_F; ++jl) {
            const int gj = (jc * CHUNK_F + jl) & 7;   // compile-time after unroll
            float r[G];
            #pragma unroll
            for (int g = 0; g < G; ++g) r[g] = lt[jl * (G * D) + g * D];
            // acc += sum_gi x[b,j,gi,d] * w[gi][gj] * S[gi][gj]   (pairs i<j)
            #pragma unroll
            for (int gi = 0; gi < G; ++gi)
                acc = fmaf(r[gi], wsh[gi * G + gj] * S[gi][gj], acc);
            // then include field j into the prefix sums: S[gj][g] += x[b,j,g,d]
            #pragma unroll
            for (int g = 0; g < G; ++g) S[gj][g] += r[g];
        }
        __syncthreads();                   // buffer jc&1 free for reuse next iteration
    }

    out[(size_t)(b0 + q) * D + d] = acc;
}

extern "C" void kernel_launch(void* const* d_in, const int* in_sizes, int n_in,
                              void* d_out, int out_size, void* d_ws, size_t ws_size,
                              hipStream_t stream) {
    const float* inputs = (const float*)d_in[0];   // (B, F, G, D) fp32
    const float* corr   = (const float*)d_in[1];   // (G, G, 1)  fp32
    float*       out    = (float*)d_out;           // (B, D)     fp32
    (void)in_sizes; (void)n_in; (void)out_size; (void)d_ws; (void)ws_size;

    dim3 grid(B_TOT / NB);   // 512 blocks
    dim3 block(THREADS);     // 256 threads = 8 waves
    GwPFM_pairsum_kernel<<<grid, block, 0, stream>>>(inputs, corr, out);
}